// MinimizeEnergy_73495480369803
// MI455X (gfx1250) — compile-verified
//
#include <hip/hip_runtime.h>
#include <math.h>

typedef __attribute__((ext_vector_type(2))) float v2f;
typedef __attribute__((ext_vector_type(8))) float v8f;

#define PI_F 3.14159265358979323846f

// Exact f32 sum of x over all 32 lanes of a wave using the CDNA5 matrix pipe.
// A: lane L puts x in (M=L%16, K = L<16 ? 0 : 2), zero elsewhere.
// B: all-ones 4x16.  D(m,n) = x_m + x_{m+16}  (= s_m) for every column n.
// C/D layout: lane l<16, vgpr r -> D(r,l) = s_r ; lane l>=16 -> D(r+8, l-16) = s_{r+8}.
// So lanes 0-15 can sum s_0..s_7 from their 8 VGPRs, lanes 16-31 sum s_8..s_15,
// and one shfl_xor(16) finishes the reduction. All lanes must be active (EXEC=~0).
__device__ __forceinline__ float wave_sum32(float x) {
    v2f a; a[0] = x;    a[1] = 0.0f;
    v2f b; b[0] = 1.0f; b[1] = 1.0f;
    v8f c = {};
    c = __builtin_amdgcn_wmma_f32_16x16x4_f32(false, a, false, b, (short)0, c, false, false);
    float t = ((c[0] + c[1]) + (c[2] + c[3])) + ((c[4] + c[5]) + (c[6] + c[7]));
    t += __shfl_xor(t, 16, 32);
    return t;   // every lane holds the full 32-lane sum
}

__global__ void init_acc(double* acc) {
    if (threadIdx.x < 3) acc[threadIdx.x] = 0.0;
}

__global__ void bond_kernel(const float* __restrict__ pos,
                            const int* __restrict__ idcs,
                            const float* __restrict__ eq,
                            const float* __restrict__ tol,
                            double* __restrict__ acc, int n) {
    float s = 0.0f;
    int stride = gridDim.x * blockDim.x;
    for (int i = blockIdx.x * blockDim.x + threadIdx.x; i < n; i += stride) {
        int a0 = idcs[2 * i];
        int a1 = idcs[2 * i + 1];
        float dx = pos[3 * a0 + 0] - pos[3 * a1 + 0];
        float dy = pos[3 * a0 + 1] - pos[3 * a1 + 1];
        float dz = pos[3 * a0 + 2] - pos[3 * a1 + 2];
        float d  = sqrtf(dx * dx + dy * dy + dz * dz);
        float r  = d - eq[i];
        float t  = tol[i];
        s += fmaxf(r * r - t * t, 0.0f);
    }
    float w = wave_sum32(s);
    if ((threadIdx.x & 31) == 0) atomicAdd(acc + 0, (double)w);
}

__global__ void angle_kernel(const float* __restrict__ pos,
                             const int* __restrict__ idcs,
                             const float* __restrict__ eq,
                             const float* __restrict__ tol,
                             double* __restrict__ acc, int n) {
    float s = 0.0f;
    int stride = gridDim.x * blockDim.x;
    for (int i = blockIdx.x * blockDim.x + threadIdx.x; i < n; i += stride) {
        int ai = idcs[3 * i + 0];
        int aj = idcs[3 * i + 1];
        int ak = idcs[3 * i + 2];
        float jx = pos[3 * aj + 0], jy = pos[3 * aj + 1], jz = pos[3 * aj + 2];
        float b0x = pos[3 * ai + 0] - jx;
        float b0y = pos[3 * ai + 1] - jy;
        float b0z = pos[3 * ai + 2] - jz;
        float b1x = pos[3 * ak + 0] - jx;
        float b1y = pos[3 * ak + 1] - jy;
        float b1z = pos[3 * ak + 2] - jz;
        float n0 = sqrtf(b0x * b0x + b0y * b0y + b0z * b0z);
        float n1 = sqrtf(b1x * b1x + b1y * b1y + b1z * b1z);
        float c  = (b0x * b1x + b0y * b1y + b0z * b1z) / (n0 * n1);
        c = fminf(fmaxf(c, -1.0f + 1e-7f), 1.0f - 1e-7f);
        float a = acosf(c);
        float r = a - eq[i];
        float t = tol[i];
        s += fmaxf(r * r - t * t, 0.0f);
    }
    float w = wave_sum32(s);
    if ((threadIdx.x & 31) == 0) atomicAdd(acc + 1, (double)w);
}

__global__ void dih_kernel(const float* __restrict__ pos,
                           const int* __restrict__ idcs,
                           const float* __restrict__ eq,
                           double* __restrict__ acc, int n) {
    float s = 0.0f;
    int stride = gridDim.x * blockDim.x;
    for (int i = blockIdx.x * blockDim.x + threadIdx.x; i < n; i += stride) {
        int a0 = idcs[4 * i + 0];
        int a1 = idcs[4 * i + 1];
        int a2 = idcs[4 * i + 2];
        int a3 = idcs[4 * i + 3];
        float p0x = pos[3 * a0 + 0], p0y = pos[3 * a0 + 1], p0z = pos[3 * a0 + 2];
        float p1x = pos[3 * a1 + 0], p1y = pos[3 * a1 + 1], p1z = pos[3 * a1 + 2];
        float p2x = pos[3 * a2 + 0], p2y = pos[3 * a2 + 1], p2z = pos[3 * a2 + 2];
        float p3x = pos[3 * a3 + 0], p3y = pos[3 * a3 + 1], p3z = pos[3 * a3 + 2];
        float b0x = p0x - p1x, b0y = p0y - p1y, b0z = p0z - p1z;
        float b1x = p2x - p1x, b1y = p2y - p1y, b1z = p2z - p1z;
        float b2x = p3x - p2x, b2y = p3y - p2y, b2z = p3z - p2z;
        float invn = 1.0f / sqrtf(b1x * b1x + b1y * b1y + b1z * b1z);
        b1x *= invn; b1y *= invn; b1z *= invn;
        float d0 = b0x * b1x + b0y * b1y + b0z * b1z;
        float vx = b0x - d0 * b1x, vy = b0y - d0 * b1y, vz = b0z - d0 * b1z;
        float d2 = b2x * b1x + b2y * b1y + b2z * b1z;
        float wx = b2x - d2 * b1x, wy = b2y - d2 * b1y, wz = b2z - d2 * b1z;
        float x = vx * wx + vy * wy + vz * wz;
        // cross(b1, v)
        float cx = b1y * vz - b1z * vy;
        float cy = b1z * vx - b1x * vz;
        float cz = b1x * vy - b1y * vx;
        float y = cx * wx + cy * wy + cz * wz;
        float dd = atan2f(y, x) - eq[i];
        s += 2.0f + cosf(dd - PI_F) + sinf(dd - 0.5f * PI_F);
    }
    float w = wave_sum32(s);
    if ((threadIdx.x & 31) == 0) atomicAdd(acc + 2, (double)w);
}

__global__ void finalize_kernel(const double* __restrict__ acc,
                                float* __restrict__ out,
                                int nb, int na, int nd) {
    if (threadIdx.x == 0 && blockIdx.x == 0) {
        double be = 1000.0 * (acc[0] / (double)nb);
        double ae = 150.0  * (acc[1] / (double)na);
        double de = acc[2] / (double)nd;
        out[0] = (float)(be + ae + de);
        out[1] = (float)be;
        out[2] = (float)ae;
        out[3] = (float)de;
    }
}

extern "C" void kernel_launch(void* const* d_in, const int* in_sizes, int n_in,
                              void* d_out, int out_size, void* d_ws, size_t ws_size,
                              hipStream_t stream) {
    const float* pos  = (const float*)d_in[0];
    const int*   bidx = (const int*)  d_in[1];
    const float* beq  = (const float*)d_in[2];
    const float* btol = (const float*)d_in[3];
    const int*   aidx = (const int*)  d_in[4];
    const float* aeq  = (const float*)d_in[5];
    const float* atol = (const float*)d_in[6];
    const int*   didx = (const int*)  d_in[7];
    const float* deq  = (const float*)d_in[8];
    int nb = in_sizes[2];   // bond_eq_val count
    int na = in_sizes[5];   // angle_eq_val count
    int nd = in_sizes[8];   // dih_eq_val count

    double* acc = (double*)d_ws;
    float*  out = (float*)d_out;

    init_acc<<<1, 32, 0, stream>>>(acc);

    const int threads = 256;
    const int blocks  = 2048;
    bond_kernel <<<blocks, threads, 0, stream>>>(pos, bidx, beq, btol, acc, nb);
    angle_kernel<<<blocks, threads, 0, stream>>>(pos, aidx, aeq, atol, acc, na);
    dih_kernel  <<<blocks, threads, 0, stream>>>(pos, didx, deq,       acc, nd);

    finalize_kernel<<<1, 32, 0, stream>>>(acc, out, nb, na, nd);
}